// OrthoLoss_52518860096014
// MI455X (gfx1250) — compile-verified
//
#include <hip/hip_runtime.h>
#include <math.h>

#define BLOCK 256
#define ELEMS 4

typedef __attribute__((ext_vector_type(2))) float v2f;
typedef __attribute__((ext_vector_type(8))) float v8f;

// ---- fast hardware math (1-instruction v_rcp/v_sqrt/v_cos) -----------------
__device__ __forceinline__ float frcp(float x) {
#if __has_builtin(__builtin_amdgcn_rcpf)
  return __builtin_amdgcn_rcpf(x);
#else
  return 1.0f / x;
#endif
}
__device__ __forceinline__ float fsqrt_hw(float x) {
#if __has_builtin(__builtin_amdgcn_sqrtf)
  return __builtin_amdgcn_sqrtf(x);
#else
  return sqrtf(x);
#endif
}
// cos(2*pi*x); v_cos_f32 operand is in revolutions, x in [0,1] here so no fract
__device__ __forceinline__ float fcos2pi(float x) {
#if __has_builtin(__builtin_amdgcn_cosf)
  return __builtin_amdgcn_cosf(x);
#else
  return cosf(6.283185307179586f * x);
#endif
}

// ---------------------------------------------------------------------------
// Wave-level sum of (ps0 + ps1) across all 32 lanes with one
// V_WMMA_F32_16X16X4_F32: A (16x4 f32 = 2 VGPRs) holds the 64 partials,
// B = all-ones (layout independent), D rows are K-sums. Summing D's 8 VGPRs
// yields half-wave totals in lanes 0-15 / 16-31; one shfl_xor(16) finishes.
// Requires EXEC all-ones (call from fully-active code only).
// ---------------------------------------------------------------------------
__device__ __forceinline__ float wave_sum_wmma(float ps0, float ps1) {
#if __has_builtin(__builtin_amdgcn_wmma_f32_16x16x4_f32)
  v2f A;  A[0] = ps0;  A[1] = ps1;
  v2f Bn; Bn[0] = 1.0f; Bn[1] = 1.0f;
  v8f C = {};
  v8f D = __builtin_amdgcn_wmma_f32_16x16x4_f32(
      false, A, false, Bn, (short)0, C, false, false);
  float t = D[0] + D[1] + D[2] + D[3] + D[4] + D[5] + D[6] + D[7];
  t += __shfl_xor(t, 16, 32);
  return t;                        // every lane holds the wave total
#else
  float t = ps0 + ps1;
  #pragma unroll
  for (int off = 16; off >= 1; off >>= 1) t += __shfl_xor(t, off, 32);
  return t;
#endif
}

// ---------------------------------------------------------------------------
// Per-matrix loss: closed-form eigenvalues of G = W^T W; smallest eigenvalue
// refined via det(G)/(l1*l2) (det(G)=det(W)^2 is relatively accurate), since
// the 1/s^2 term is heavy-tailed for near-singular W.
// ---------------------------------------------------------------------------
__device__ __forceinline__ float elem_loss(const float4* __restrict__ tp,
                                           long long idx) {
  float4 r0 = tp[3 * idx + 0];
  float4 r1 = tp[3 * idx + 1];
  float4 r2 = tp[3 * idx + 2];
  float w00 = r0.x, w01 = r0.y, w02 = r0.z;
  float w10 = r1.x, w11 = r1.y, w12 = r1.z;
  float w20 = r2.x, w21 = r2.y, w22 = r2.z;

  // Gram matrix G = W^T W (a,b,c diag; d=G01, e=G02, f=G12)
  float a = fmaf(w20, w20, fmaf(w10, w10, w00 * w00));
  float b = fmaf(w21, w21, fmaf(w11, w11, w01 * w01));
  float c = fmaf(w22, w22, fmaf(w12, w12, w02 * w02));
  float d = fmaf(w20, w21, fmaf(w10, w11, w00 * w01));
  float e = fmaf(w20, w22, fmaf(w10, w12, w00 * w02));
  float f = fmaf(w21, w22, fmaf(w11, w12, w01 * w02));

  float detW = w00 * (w11 * w22 - w12 * w21)
             - w01 * (w10 * w22 - w12 * w20)
             + w02 * (w10 * w21 - w11 * w20);
  float detG = detW * detW;

  const float third = 0.33333333333333333f;
  float q  = (a + b + c) * third;
  float aq = a - q, bq = b - q, cq = c - q;
  float p2 = aq * aq + bq * bq + cq * cq + 2.0f * (d * d + e * e + f * f);
  float p  = fsqrt_hw(p2 * (1.0f / 6.0f));
  float ip = frcp(fmaxf(p, 1e-30f));

  // r = det((G - qI)/p) / 2, clamped to [-1, 1]
  float detC = aq * (bq * cq - f * f)
             - d  * (d * cq - f * e)
             + e  * (d * f - bq * e);
  float r = 0.5f * detC * (ip * ip * ip);
  r = fminf(1.0f, fmaxf(-1.0f, r));

  // phi = acos(r)/3, expressed in revolutions for v_cos_f32
  const float inv6pi = 0.05305164769729845f;  // 1/(6*pi) = (1/3)*(1/2pi)
  float phin = acosf(r) * inv6pi;             // in [0, 1/6]
  float twop = 2.0f * p;
  float l1 = q + twop * fcos2pi(phin);                 // largest
  float l3 = q + twop * fcos2pi(phin + third);         // smallest
  float l2 = 3.0f * q - l1 - l3;

  l1 = fmaxf(l1, 0.0f);
  l2 = fmaxf(l2, 0.0f);
  float prod = l1 * l2;
  l3 = (prod > 1e-28f) ? (detG * frcp(prod)) : fmaxf(l3, 0.0f);

  const float eps = 1e-6f;
  float s1 = fsqrt_hw(l1) + eps;
  float s2 = fsqrt_hw(l2) + eps;
  float s3 = fsqrt_hw(l3) + eps;
  float q1 = s1 * s1, q2 = s2 * s2, q3 = s3 * s3;
  return (q1 + frcp(q1)) + (q2 + frcp(q2)) + (q3 + frcp(q3));
}

// ---------------------------------------------------------------------------
// Kernel 1: grid-stride partial sums, WMMA wave reduce, LDS block reduce,
// one partial per block (deterministic; no float atomics).
// ---------------------------------------------------------------------------
__global__ void OrthoLoss_partial_kernel(const float4* __restrict__ tp,
                                         float* __restrict__ partial,
                                         long long Bn) {
  __shared__ float lds[BLOCK / 32];
  long long tid    = (long long)blockIdx.x * BLOCK + threadIdx.x;
  long long stride = (long long)gridDim.x * BLOCK;

  float ps0 = 0.0f, ps1 = 0.0f;
  #pragma unroll
  for (int i = 0; i < ELEMS; ++i) {
    long long idx = tid + (long long)i * stride;
    if (i + 1 < ELEMS) {
      // speculative prefetch of next tile -> global_prefetch_b8
      __builtin_prefetch((const void*)(tp + 3 * (idx + stride)), 0, 0);
    }
    float v = 0.0f;
    if (idx < Bn) v = elem_loss(tp, idx);
    if (i & 1) ps1 += v; else ps0 += v;
  }

  float w = wave_sum_wmma(ps0, ps1);   // all threads active here
  int lane = threadIdx.x & 31;
  int wid  = threadIdx.x >> 5;
  if (lane == 0) lds[wid] = w;
  __syncthreads();
  if (threadIdx.x == 0) {
    float s = 0.0f;
    #pragma unroll
    for (int i = 0; i < BLOCK / 32; ++i) s += lds[i];
    partial[blockIdx.x] = s;
  }
}

// ---------------------------------------------------------------------------
// Kernel 2: single block reduces the block partials; writes mean - 6.
// ---------------------------------------------------------------------------
__global__ void OrthoLoss_finish_kernel(const float* __restrict__ partial,
                                        int n, float* __restrict__ out,
                                        float invB) {
  __shared__ float lds[BLOCK / 32];
  float ps0 = 0.0f, ps1 = 0.0f;
  int iter = 0;
  for (int i = threadIdx.x; i < n; i += BLOCK, ++iter) {
    if (iter & 1) ps1 += partial[i]; else ps0 += partial[i];
  }
  float w = wave_sum_wmma(ps0, ps1);
  int lane = threadIdx.x & 31;
  int wid  = threadIdx.x >> 5;
  if (lane == 0) lds[wid] = w;
  __syncthreads();
  if (threadIdx.x == 0) {
    float s = 0.0f;
    #pragma unroll
    for (int i = 0; i < BLOCK / 32; ++i) s += lds[i];
    out[0] = s * invB - 6.0f;
  }
}

extern "C" void kernel_launch(void* const* d_in, const int* in_sizes, int n_in,
                              void* d_out, int out_size, void* d_ws, size_t ws_size,
                              hipStream_t stream) {
  const float4* tp = (const float4*)d_in[0];
  long long Bn = (long long)in_sizes[0] / 12;      // B matrices of 3x4 f32
  if (Bn < 1) Bn = 1;

  long long perBlock = (long long)BLOCK * ELEMS;   // 1024 matrices per block
  int nBlocks = (int)((Bn + perBlock - 1) / perBlock);
  if (nBlocks < 1) nBlocks = 1;                    // B=2^21 -> 2048 blocks

  float* partial = (float*)d_ws;                   // nBlocks floats of scratch

  OrthoLoss_partial_kernel<<<nBlocks, BLOCK, 0, stream>>>(tp, partial, Bn);
  OrthoLoss_finish_kernel<<<1, BLOCK, 0, stream>>>(partial, nBlocks,
                                                   (float*)d_out,
                                                   1.0f / (float)Bn);
}